// NeuralODE_60473139528046
// MI455X (gfx1250) — compile-verified
//
#include <hip/hip_runtime.h>
#include <math.h>

typedef __attribute__((ext_vector_type(2))) float v2f;
typedef __attribute__((ext_vector_type(8))) float v8f;

// ---------------- problem constants (match reference) ----------------
#define DIM   32     // state dim D
#define WID   128    // hidden width W
#define NSUB  4      // fixed Tsit5 substeps per save interval

// ---------------- LDS layout (floats), dynamic shared mem ------------
#define ACT_S 132            // padded stride for [16][128] activation tiles
#define Y_S   36             // padded stride for [16][32] state tiles
#define OFF_WIN   0
#define SZ_WIN    (8*8*32*2)            // 4096  : W_in  B-fragments
#define OFF_WH    (OFF_WIN + SZ_WIN)
#define SZ_WH     (2*8*32*32*2)         // 32768 : W_h   B-fragments (2 layers)
#define OFF_WOUT  (OFF_WH + SZ_WH)
#define SZ_WOUT   (2*32*32*2)           // 4096  : W_out B-fragments
#define OFF_BIN   (OFF_WOUT + SZ_WOUT)  // 128
#define OFF_BH    (OFF_BIN + 128)       // 256
#define OFF_BOUT  (OFF_BH + 256)        // 32
#define OFF_ACT0  (OFF_BOUT + 32)
#define OFF_ACT1  (OFF_ACT0 + 16*ACT_S)
#define OFF_YST   (OFF_ACT1 + 16*ACT_S)
#define OFF_Y     (OFF_YST + 16*Y_S)
#define OFF_K     (OFF_Y + 16*Y_S)      // 6 stage slots [6][16][Y_S]
#define SMEM_FLOATS (OFF_K + 6*16*Y_S)  // = 50208 floats = 200832 bytes < 320KB

// ---------------- Tsit5 coefficients ---------------------------------
__device__ __constant__ float cA2[1] = {0.161f};
__device__ __constant__ float cA3[2] = {-0.008480655492356989f, 0.335480655492357f};
__device__ __constant__ float cA4[3] = {2.8971530571054935f, -6.359448489975075f, 4.3622954328695815f};
__device__ __constant__ float cA5[4] = {5.325864828439257f, -11.748883564062828f, 7.4955393428898365f, -0.09249506636175525f};
__device__ __constant__ float cA6[5] = {5.86145544294642f, -12.92096931784711f, 8.159367898576159f, -0.071584973281401f, -0.028269050394068383f};
__device__ __constant__ float cB[6]  = {0.09646076681806523f, 0.01f, 0.4798896504144996f,
                                        1.379008574103742f, -3.290069515436081f, 2.324710524099774f};

#define LOG2E_F 1.4426950408889634f
#define LN2_F   0.6931471805599453f

__device__ __forceinline__ float softplus_f(float x) {
    // stable softplus, raw hardware trans ops (no subnormal guards needed):
    //   max(x,0) + ln2 * log2(1 + exp2(-|x|*log2e))
    // exp2 arg <= 0 (no overflow); log2 arg in [1,2] (never subnormal).
    float t = __builtin_amdgcn_exp2f(-LOG2E_F * fabsf(x));
    return fmaf(LN2_F, __builtin_amdgcn_logf(1.0f + t), fmaxf(x, 0.0f));
}

__device__ __forceinline__ v8f wmma_f32(v2f a, v2f b, v8f c) {
    return __builtin_amdgcn_wmma_f32_16x16x4_f32(false, a, false, b, (short)0, c,
                                                 false, false);
}

// One MLP evaluation k = f(yst) for a 16-row batch tile.
// 8 waves split the 128-wide layers (one 16-wide N-tile each); waves 0,1 do
// the 32-wide output layer. Activations bounce through LDS between layers.
__device__ __forceinline__ void mlp_eval(float* sm, const float* yst, float* kout,
                                         int wave, int lane) {
    float* act0 = sm + OFF_ACT0;
    float* act1 = sm + OFF_ACT1;
    const float* win  = sm + OFF_WIN;
    const float* wh   = sm + OFF_WH;
    const float* wout = sm + OFF_WOUT;
    const float* bin  = sm + OFF_BIN;
    const float* bh   = sm + OFF_BH;
    const float* bout = sm + OFF_BOUT;

    const int l15   = lane & 15;
    const int koff  = (lane < 16) ? 0 : 2;   // A-fragment K sub-offset
    const int rbase = (lane < 16) ? 0 : 8;   // D-fragment row base

    // ---- input layer: [16x32] @ W_in^T -> [16x128], softplus ----
    {
        v2f a[DIM / 4];
#pragma unroll
        for (int t = 0; t < DIM / 4; ++t)
            a[t] = *(const v2f*)&yst[l15 * Y_S + 4 * t + koff];
        v8f acc = {};
        const int m = wave;
#pragma unroll
        for (int t = 0; t < DIM / 4; ++t) {
            v2f b = *(const v2f*)&win[(((m * 8) + t) * 32 + lane) * 2];
            acc = wmma_f32(a[t], b, acc);
        }
        const float bv  = bin[16 * m + l15];
        const int   col = 16 * m + l15;
#pragma unroll
        for (int r = 0; r < 8; ++r)
            act0[(rbase + r) * ACT_S + col] = softplus_f(acc[r] + bv);
    }
    __syncthreads();

    // ---- two hidden layers: [16x128] @ W_h[i]^T -> [16x128], softplus ----
#pragma unroll
    for (int li = 0; li < 2; ++li) {
        const float* src = (li == 0) ? act0 : act1;
        float*       dst = (li == 0) ? act1 : act0;
        v2f a[WID / 4];
#pragma unroll
        for (int t = 0; t < WID / 4; ++t)
            a[t] = *(const v2f*)&src[l15 * ACT_S + 4 * t + koff];
        v8f acc = {};
        const int m = wave;
        const float* wf = wh + li * (8 * 32 * 64);
#pragma unroll
        for (int t = 0; t < WID / 4; ++t) {
            v2f b = *(const v2f*)&wf[(((m * 32) + t) * 32 + lane) * 2];
            acc = wmma_f32(a[t], b, acc);
        }
        const float bv  = bh[li * WID + 16 * m + l15];
        const int   col = 16 * m + l15;
#pragma unroll
        for (int r = 0; r < 8; ++r)
            dst[(rbase + r) * ACT_S + col] = softplus_f(acc[r] + bv);
        __syncthreads();
    }

    // ---- output layer: [16x128] @ W_out^T -> [16x32] (waves 0,1) ----
    if (wave < 2) {
        v2f a[WID / 4];
#pragma unroll
        for (int t = 0; t < WID / 4; ++t)
            a[t] = *(const v2f*)&act0[l15 * ACT_S + 4 * t + koff];
        v8f acc = {};
        const int m = wave;
#pragma unroll
        for (int t = 0; t < WID / 4; ++t) {
            v2f b = *(const v2f*)&wout[(((m * 32) + t) * 32 + lane) * 2];
            acc = wmma_f32(a[t], b, acc);
        }
        const float bv  = bout[16 * m + l15];
        const int   col = 16 * m + l15;
#pragma unroll
        for (int r = 0; r < 8; ++r)
            kout[(rbase + r) * Y_S + col] = acc[r] + bv;
    }
    __syncthreads();
}

// yst = y + h * sum_j c[j] * k_j   over the [16][32] tile (stride Y_S)
template <int NT>
__device__ __forceinline__ void stage_combine(float* yst, const float* ybuf,
                                              const float* kbuf, const float* c,
                                              float h, int tid) {
    for (int e = tid; e < 16 * DIM; e += 256) {
        const int row = e >> 5, col = e & 31, idx = row * Y_S + col;
        float acc = ybuf[idx];
#pragma unroll
        for (int j = 0; j < NT; ++j)
            acc = fmaf(h * c[j], kbuf[j * (16 * Y_S) + idx], acc);
        yst[idx] = acc;
    }
    __syncthreads();
}

__global__ void __launch_bounds__(256, 1)
node_tsit5_kernel(const float* __restrict__ ts, const float* __restrict__ y0,
                  const float* __restrict__ W_in, const float* __restrict__ b_in,
                  const float* __restrict__ W_h, const float* __restrict__ b_h,
                  const float* __restrict__ W_out, const float* __restrict__ b_out,
                  float* __restrict__ out, int T) {
    extern __shared__ float sm[];
    const int tid  = threadIdx.x;
    const int lane = tid & 31;
    const int wave = tid >> 5;
    const int tb   = blockIdx.x * 16;  // batch tile base row

    // ---------- load weights, pre-swizzled into WMMA B-fragment layout ----------
    // B-fragment (K=4,N=16): lane<16 holds W[n=16m+l][4t+{0,1}], lane>=16 holds +{2,3}
    for (int q = tid; q < SZ_WIN; q += 256) {          // W_in [128][32]
        int m = q >> 9, t = (q >> 6) & 7, e = q & 63;
        int ln = e >> 1, j = e & 1;
        int n = 16 * m + (ln & 15);
        int k = 4 * t + ((ln < 16) ? 0 : 2) + j;
        sm[OFF_WIN + q] = W_in[n * DIM + k];
    }
    for (int q = tid; q < SZ_WH; q += 256) {           // W_h [2][128][128]
        int i = q >> 14, r = q & 16383;
        int m = r >> 11, t = (r >> 6) & 31, e = r & 63;
        int ln = e >> 1, j = e & 1;
        int n = 16 * m + (ln & 15);
        int k = 4 * t + ((ln < 16) ? 0 : 2) + j;
        sm[OFF_WH + q] = W_h[i * WID * WID + n * WID + k];
    }
    for (int q = tid; q < SZ_WOUT; q += 256) {         // W_out [32][128]
        int m = q >> 11, t = (q >> 6) & 31, e = q & 63;
        int ln = e >> 1, j = e & 1;
        int n = 16 * m + (ln & 15);
        int k = 4 * t + ((ln < 16) ? 0 : 2) + j;
        sm[OFF_WOUT + q] = W_out[n * WID + k];
    }
    for (int q = tid; q < 128; q += 256) sm[OFF_BIN + q]  = b_in[q];
    for (int q = tid; q < 256; q += 256) sm[OFF_BH + q]   = b_h[q];
    for (int q = tid; q < 32;  q += 256) sm[OFF_BOUT + q] = b_out[q];

    // ---------- initial state + t0 snapshot ----------
    float* ybuf = sm + OFF_Y;
    float* yst  = sm + OFF_YST;
    float* kbuf = sm + OFF_K;
    for (int e = tid; e < 16 * DIM; e += 256) {
        int row = e >> 5, col = e & 31;
        float v = y0[(size_t)(tb + row) * DIM + col];
        ybuf[row * Y_S + col] = v;
        out[(size_t)(tb + row) * T * DIM + col] = v;  // ys[:,0,:] = y0
    }
    __syncthreads();

    // ---------- time integration ----------
#pragma unroll 1
    for (int iv = 0; iv < T - 1; ++iv) {
        const float dt = ts[iv + 1] - ts[iv];
        const float h  = dt * (1.0f / NSUB);
#pragma unroll 1
        for (int sub = 0; sub < NSUB; ++sub) {
            stage_combine<0>(yst, ybuf, kbuf, cB, h, tid);          // yst = y
            mlp_eval(sm, yst, kbuf + 0 * (16 * Y_S), wave, lane);   // k1
            stage_combine<1>(yst, ybuf, kbuf, cA2, h, tid);
            mlp_eval(sm, yst, kbuf + 1 * (16 * Y_S), wave, lane);   // k2
            stage_combine<2>(yst, ybuf, kbuf, cA3, h, tid);
            mlp_eval(sm, yst, kbuf + 2 * (16 * Y_S), wave, lane);   // k3
            stage_combine<3>(yst, ybuf, kbuf, cA4, h, tid);
            mlp_eval(sm, yst, kbuf + 3 * (16 * Y_S), wave, lane);   // k4
            stage_combine<4>(yst, ybuf, kbuf, cA5, h, tid);
            mlp_eval(sm, yst, kbuf + 4 * (16 * Y_S), wave, lane);   // k5
            stage_combine<5>(yst, ybuf, kbuf, cA6, h, tid);
            mlp_eval(sm, yst, kbuf + 5 * (16 * Y_S), wave, lane);   // k6
            // y += h * sum_j B_j k_j
            for (int e = tid; e < 16 * DIM; e += 256) {
                int row = e >> 5, col = e & 31, idx = row * Y_S + col;
                float acc = ybuf[idx];
#pragma unroll
                for (int j = 0; j < 6; ++j)
                    acc = fmaf(h * cB[j], kbuf[j * (16 * Y_S) + idx], acc);
                ybuf[idx] = acc;
            }
            __syncthreads();
        }
        // save snapshot at ts[iv+1]  (coalesced: 32 consecutive floats per row)
        for (int e = tid; e < 16 * DIM; e += 256) {
            int row = e >> 5, col = e & 31;
            out[(size_t)(tb + row) * T * DIM + (size_t)(iv + 1) * DIM + col] =
                ybuf[row * Y_S + col];
        }
    }
}

extern "C" void kernel_launch(void* const* d_in, const int* in_sizes, int n_in,
                              void* d_out, int out_size, void* d_ws, size_t ws_size,
                              hipStream_t stream) {
    const float* ts    = (const float*)d_in[0];
    const float* y0    = (const float*)d_in[1];
    const float* W_in  = (const float*)d_in[2];
    const float* b_in  = (const float*)d_in[3];
    const float* W_h   = (const float*)d_in[4];
    const float* b_h   = (const float*)d_in[5];
    const float* W_out = (const float*)d_in[6];
    const float* b_out = (const float*)d_in[7];
    float* out = (float*)d_out;

    const int T = in_sizes[0];           // 128 save times
    const int B = in_sizes[1] / DIM;     // 4096 trajectories
    const size_t shmem = (size_t)SMEM_FLOATS * sizeof(float);  // ~196 KB < 320 KB

    hipFuncSetAttribute((const void*)node_tsit5_kernel,
                        hipFuncAttributeMaxDynamicSharedMemorySize, (int)shmem);
    node_tsit5_kernel<<<B / 16, 256, shmem, stream>>>(
        ts, y0, W_in, b_in, W_h, b_h, W_out, b_out, out, T);
}